// RelativePositionBias2D_76794015252602
// MI455X (gfx1250) — compile-verified
//
#include <hip/hip_runtime.h>

// CDNA5 (gfx1250) fused iRPE bias kernel:
//   lookup = Q @ W  (per b,h; 196x64 @ 64x169, fp32 WMMA 16x16x4) -> LDS
//   out[i,j] = lookup[i, rp_bucket[i,j]]  (non-temporal streamed stores)

typedef float v2f __attribute__((ext_vector_type(2)));
typedef float v8f __attribute__((ext_vector_type(8)));

#define LTOK 196   // tokens (E*E)
#define DDIM 64    // head dim
#define KBUK 169   // buckets
#define KPAD 176   // buckets padded to 11*16
#define MT   13    // ceil(196/16)
#define NT   11    // 176/16

__global__ __launch_bounds__(256)
void irpe_fused_kernel(const float* __restrict__ q,     // (B,H,L,D)
                       const float* __restrict__ w,     // (H,D,K)
                       const int*   __restrict__ rp,    // (L,L)
                       float*       __restrict__ out)   // (B,H,L,L)
{
    extern __shared__ float lookup[];       // LTOK * KPAD floats = 137,984 B

    const int bh   = blockIdx.x;            // b*H + h
    const int h    = bh & 7;                // H == 8
    const int tid  = threadIdx.x;
    const int lane = tid & 31;
    const int wave = tid >> 5;              // 0..7

    const float* qbase = q + (size_t)bh * (LTOK * DDIM);
    const float* wbase = w + (size_t)h  * (DDIM * KBUK);

    const int mrow  = lane & 15;            // M (for A) / N (for B) within tile
    const int halfK = (lane >> 4) * 2;      // lanes 16-31 hold K+2 per ISA layout
    const int mhi   = (lane >> 4) * 8;      // C/D layout: lanes 16-31 -> rows M+8

    // ---------------- GEMM phase: lookup(196x169) -> LDS ----------------
    // Waves own M-tiles in stride-8; A fragments stay in VGPRs across N-tiles.
    for (int mt = wave; mt < MT; mt += 8) {
        const int m0 = mt * 16;
        int arow = m0 + mrow;
        if (arow > LTOK - 1) arow = LTOK - 1;           // clamp pad rows
        const float* qrow = qbase + arow * DDIM;

        // 16 A fragments: A[M= lane%16][K = ks*4 + halfK + {0,1}]
        v2f a[16];
        #pragma unroll
        for (int ks = 0; ks < 16; ++ks) {
            const float* p = qrow + ks * 4 + halfK;
            v2f t; t.x = p[0]; t.y = p[1];
            a[ks] = t;
        }

        for (int nt = 0; nt < NT; ++nt) {
            const int n0 = nt * 16;
            int col = n0 + mrow;
            if (col > KBUK - 1) col = KBUK - 1;         // clamp pad cols

            v8f acc = {};
            #pragma unroll
            for (int ks = 0; ks < 16; ++ks) {
                const int kd = ks * 4 + halfK;
                // B[K=kd+{0,1}][N= lane%16] from W(D x K) row-major
                v2f bf;
                bf.x = wbase[(kd + 0) * KBUK + col];
                bf.y = wbase[(kd + 1) * KBUK + col];
                // D = A x B + C  (16x16x4 fp32)
                acc = __builtin_amdgcn_wmma_f32_16x16x4_f32(
                    /*neg_a=*/false, a[ks],
                    /*neg_b=*/false, bf,
                    /*c_mod=*/(short)0, acc,
                    /*reuse_a=*/false, /*reuse_b=*/false);
            }

            // C/D layout: VGPR r holds D[m0 + mhi + r][n0 + lane%16]
            #pragma unroll
            for (int r = 0; r < 8; ++r) {
                const int rr = m0 + mhi + r;
                if (rr < LTOK)
                    lookup[rr * KPAD + n0 + mrow] = acc[r];
            }
        }
    }

    __syncthreads();

    // ---------------- Gather phase: stream 196x196 output ----------------
    float* obase = out + (size_t)bh * (LTOK * LTOK);
    for (int i = wave; i < LTOK; i += 8) {
        const float* lrow = lookup + i * KPAD;
        const int*   rrow = rp     + i * LTOK;
        float*       orow = obase  + i * LTOK;
        for (int j = lane; j < LTOK; j += 32) {
            const int k = rrow[j];                       // 0..168
            __builtin_nontemporal_store(lrow[k], orow + j);
        }
    }
}

extern "C" void kernel_launch(void* const* d_in, const int* in_sizes, int n_in,
                              void* d_out, int out_size, void* d_ws, size_t ws_size,
                              hipStream_t stream) {
    const float* q   = (const float*)d_in[0];   // (B,H,L,D) fp32
    const float* w   = (const float*)d_in[1];   // (H,D,169) fp32
    const int*   rp  = (const int*)  d_in[2];   // (L,L) int32
    float*       out = (float*)d_out;           // (B,H,L,L) fp32

    const int BH = in_sizes[0] / (LTOK * DDIM); // B*H = 1024
    const size_t lds_bytes = (size_t)LTOK * KPAD * sizeof(float); // 137,984 B

    irpe_fused_kernel<<<dim3(BH), dim3(256), lds_bytes, stream>>>(q, w, rp, out);
}